// AssociationLoss_50903952392404
// MI455X (gfx1250) — compile-verified
//
#include <hip/hip_runtime.h>
#include <math.h>

#define NPTS   8192
#define DIM    128
#define MAXID  4000
#define MARGIN 0.1f
#define BSTRIDE 132   // 16x128 tile row stride in LDS; 132 % 64 == 4 -> conflict-free ds_load_b64

typedef __attribute__((ext_vector_type(2))) float v2f;
typedef __attribute__((ext_vector_type(8))) float v8f;

// ---------------------------------------------------------------- K1: sq[i] = ||x_i||^2
__global__ void sq_norm_kernel(const float* __restrict__ x, float* __restrict__ sq) {
    int i = blockIdx.x * blockDim.x + threadIdx.x;
    if (i >= NPTS) return;
    const float4* r = (const float4*)(x + (size_t)i * DIM);
    float s = 0.f;
#pragma unroll
    for (int k = 0; k < DIM / 4; ++k) {
        float4 v = r[k];
        s += v.x * v.x + v.y * v.y + v.z * v.z + v.w * v.w;
    }
    sq[i] = s;
}

// ---------------------------------------------------------------- K2: fused Gram(WMMA f32) + hardest-pos/neg mining
// Grid: NPTS/128 blocks, 256 threads (8 waves). Wave w owns anchors [i0+16w, i0+16w+16).
// B j-tile (16x128) is double-buffered in LDS, shared by all 8 waves.
// Fragment stream from LDS is software-pipelined with a 4-fragment lookahead window.
__global__ __launch_bounds__(256)
void hard_mine_kernel(const float* __restrict__ x, const int* __restrict__ ids,
                      const float* __restrict__ sq,
                      float* __restrict__ hp_out, float* __restrict__ hn_out) {
    __shared__ float btile[2][16 * BSTRIDE];

    const int tid  = threadIdx.x;
    const int wave = tid >> 5;
    const int lane = tid & 31;
    const int half = lane >> 4;   // K/M-group select (lanes 0-15 vs 16-31)
    const int l16  = lane & 15;
    const int i0   = blockIdx.x * 128 + wave * 16;

    // Cooperative tile-load mapping: 256 threads x 8 floats = 16 rows x 128 cols.
    const int lr = tid >> 4;         // tile row 0..15
    const int lc = (tid & 15) * 8;   // tile col 0..120

    // A fragments (16x4 f32 layout): lane holds M=l16, K-base=2*half. Full K=128 resident.
    v2f afrag[32];
    const float* arow = x + (size_t)(i0 + l16) * DIM + 2 * half;
#pragma unroll
    for (int kk = 0; kk < 32; ++kk)
        afrag[kk] = *(const v2f*)(arow + 4 * kk);

    // Per C-slot (VGPR v) anchor metadata: M = v + 8*half.
    int   idI[8], clsI[8];
    float sqI[8];
#pragma unroll
    for (int v = 0; v < 8; ++v) {
        int m   = v + 8 * half;
        idI[v]  = ids[i0 + m];
        clsI[v] = idI[v] / 1000;
        sqI[v]  = sq[i0 + m];
    }

    float hp[8], hn[8];
#pragma unroll
    for (int v = 0; v < 8; ++v) { hp[v] = -__builtin_inff(); hn[v] = __builtin_inff(); }

    // Prologue: stage tile 0 into LDS buffer 0.
    {
        const float* g = x + (size_t)lr * DIM + lc;
        float4 s0 = *(const float4*)g;
        float4 s1 = *(const float4*)(g + 4);
        *(float4*)&btile[0][lr * BSTRIDE + lc]     = s0;
        *(float4*)&btile[0][lr * BSTRIDE + lc + 4] = s1;
    }
    __syncthreads();

    const int NT = NPTS / 16;
    for (int jt = 0; jt < NT; ++jt) {
        const int buf = jt & 1;
        const int j0  = jt * 16;

        // Stage next j-tile into registers (overlaps with compute below).
        float4 s0, s1;
        const bool havenext = (jt + 1) < NT;
        if (havenext) {
            const float* g = x + (size_t)(j0 + 16 + lr) * DIM + lc;
            __builtin_prefetch(g + 16 * DIM, 0, 1);   // tile t+2 -> global_prefetch
            s0 = *(const float4*)g;
            s1 = *(const float4*)(g + 4);
        }

        const int   idJ  = ids[j0 + l16];
        const int   clsJ = idJ / 1000;
        const float sqJ  = sq[j0 + l16];

        // B fragments (4x16 f32 layout) from LDS: lane holds N=l16, K-base=2*half.
        // 4-deep rotating lookahead so ds_load latency overlaps the WMMA chain.
        const float* bbase = &btile[buf][l16 * BSTRIDE + 2 * half];

        v2f cur0 = *(const v2f*)(bbase + 0);
        v2f cur1 = *(const v2f*)(bbase + 4);
        v2f cur2 = *(const v2f*)(bbase + 8);
        v2f cur3 = *(const v2f*)(bbase + 12);

        v8f acc = {0.f, 0.f, 0.f, 0.f, 0.f, 0.f, 0.f, 0.f};
#pragma unroll
        for (int kk = 0; kk < 32; kk += 4) {
            const int kn = (kk + 4) & 31;   // wraps on last step; keeps loop branch-free
            v2f nxt0 = *(const v2f*)(bbase + 4 * kn);
            v2f nxt1 = *(const v2f*)(bbase + 4 * (kn + 1));
            v2f nxt2 = *(const v2f*)(bbase + 4 * (kn + 2));
            v2f nxt3 = *(const v2f*)(bbase + 4 * (kn + 3));

            acc = __builtin_amdgcn_wmma_f32_16x16x4_f32(false, afrag[kk + 0], false, cur0,
                                                        (short)0, acc, false, false);
            acc = __builtin_amdgcn_wmma_f32_16x16x4_f32(false, afrag[kk + 1], false, cur1,
                                                        (short)0, acc, false, false);
            acc = __builtin_amdgcn_wmma_f32_16x16x4_f32(false, afrag[kk + 2], false, cur2,
                                                        (short)0, acc, false, false);
            acc = __builtin_amdgcn_wmma_f32_16x16x4_f32(false, afrag[kk + 3], false, cur3,
                                                        (short)0, acc, false, false);

            cur0 = nxt0; cur1 = nxt1; cur2 = nxt2; cur3 = nxt3;
        }

        // Epilogue in registers: d(i,j) + masks, update running extrema.
#pragma unroll
        for (int v = 0; v < 8; ++v) {
            float d2 = sqI[v] + sqJ - 2.0f * acc[v];
            d2 = fmaxf(d2, 0.0f);
            float d = sqrtf(d2);
            bool same_id  = (idI[v] == idJ);
            bool same_cls = (clsI[v] == clsJ);
            hp[v] = fmaxf(hp[v], same_id ? d : -__builtin_inff());
            hn[v] = fminf(hn[v], (same_cls && !same_id) ? d : __builtin_inff());
        }

        // Publish next tile into the other buffer; one barrier per iteration.
        if (havenext) {
            *(float4*)&btile[buf ^ 1][lr * BSTRIDE + lc]     = s0;
            *(float4*)&btile[buf ^ 1][lr * BSTRIDE + lc + 4] = s1;
        }
        __syncthreads();
    }

    // Reduce over N (16 lanes per group), write per-anchor results.
#pragma unroll
    for (int v = 0; v < 8; ++v) {
        float rp = hp[v], rn = hn[v];
#pragma unroll
        for (int off = 1; off < 16; off <<= 1) {
            rp = fmaxf(rp, __shfl_xor(rp, off, 32));
            rn = fminf(rn, __shfl_xor(rn, off, 32));
        }
        if (l16 == 0) {
            int m = v + 8 * half;
            hp_out[i0 + m] = rp;
            hn_out[i0 + m] = rn;
        }
    }
}

// ---------------------------------------------------------------- K3: triplet per anchor
__global__ void triplet_kernel(const float* __restrict__ hp, const float* __restrict__ hn,
                               float* __restrict__ tri) {
    int i = blockIdx.x * blockDim.x + threadIdx.x;
    if (i >= NPTS) return;
    float hnv = hn[i];
    bool has = hnv < __builtin_inff();
    tri[i] = has ? fmaxf(MARGIN + hp[i] - hnv, 0.f) : 0.f;
}

// ---------------------------------------------------------------- K4: deterministic per-id mean (no atomics)
__global__ void per_id_kernel(const int* __restrict__ ids, const float* __restrict__ tri,
                              float* __restrict__ id_loss, float* __restrict__ id_present) {
    int id = blockIdx.x * blockDim.x + threadIdx.x;
    if (id >= MAXID) return;
    float s = 0.f, c = 0.f;
    for (int i = 0; i < NPTS; ++i) {
        if (ids[i] == id) { s += tri[i]; c += 1.f; }
    }
    id_loss[id]    = (c > 0.f) ? s / c : 0.f;
    id_present[id] = (c > 0.f) ? 1.f : 0.f;
}

// ---------------------------------------------------------------- K5: final scalar
__global__ void final_kernel(const float* __restrict__ id_loss,
                             const float* __restrict__ id_present,
                             float* __restrict__ out) {
    __shared__ float ssum[256], scnt[256];
    float s = 0.f, c = 0.f;
    for (int t = threadIdx.x; t < MAXID; t += 256) { s += id_loss[t]; c += id_present[t]; }
    ssum[threadIdx.x] = s; scnt[threadIdx.x] = c;
    __syncthreads();
    for (int o = 128; o > 0; o >>= 1) {
        if ((int)threadIdx.x < o) {
            ssum[threadIdx.x] += ssum[threadIdx.x + o];
            scnt[threadIdx.x] += scnt[threadIdx.x + o];
        }
        __syncthreads();
    }
    if (threadIdx.x == 0) out[0] = ssum[0] / scnt[0];
}

extern "C" void kernel_launch(void* const* d_in, const int* in_sizes, int n_in,
                              void* d_out, int out_size, void* d_ws, size_t ws_size,
                              hipStream_t stream) {
    const float* x   = (const float*)d_in[0];
    const int*   ids = (const int*)d_in[1];
    float* out = (float*)d_out;

    float* ws      = (float*)d_ws;
    float* hp      = ws;               // NPTS
    float* hn      = ws + NPTS;        // NPTS
    float* sq      = ws + 2 * NPTS;    // NPTS
    float* tri     = ws + 3 * NPTS;    // NPTS
    float* id_loss = ws + 4 * NPTS;    // MAXID
    float* id_pres = id_loss + MAXID;  // MAXID

    sq_norm_kernel<<<NPTS / 256, 256, 0, stream>>>(x, sq);
    hard_mine_kernel<<<NPTS / 128, 256, 0, stream>>>(x, ids, sq, hp, hn);
    triplet_kernel<<<NPTS / 256, 256, 0, stream>>>(hp, hn, tri);
    per_id_kernel<<<(MAXID + 255) / 256, 256, 0, stream>>>(ids, tri, id_loss, id_pres);
    final_kernel<<<1, 256, 0, stream>>>(id_loss, id_pres, out);
}